// gbuilder_4870492914218
// MI455X (gfx1250) — compile-verified
//
#include <hip/hip_runtime.h>

typedef __attribute__((ext_vector_type(2))) float v2f;
typedef __attribute__((ext_vector_type(8))) float v8f;

#define GS    512
#define TWO_P 20
#define P     10
#define FREE  30
#define OUTC  (GS + P + FREE)   // 552 output channels per row

// ---------------------------------------------------------------------------
// Gaussian distance tile via V_WMMA_F32_16X16X4_F32.
// dsq[i,j] = n_i + n_j - 2 x4i x4j - 2 x5i x5j, encoded as a rank-4 product:
//   A[i,:] = [x4_i, x5_i, n_i, 1]   (16x4)
//   B[:,j] = [-2 x4_j, -2 x5_j, 1, n_j]  (4x16)
// One wave computes one 16x16 tile of exp(-dsq). Wave32, EXEC all ones.
// ---------------------------------------------------------------------------
__global__ __launch_bounds__(256) void gauss_tile_wmma(const float* __restrict__ x,
                                                       float* __restrict__ out) {
  const int lane = threadIdx.x & 31;
  const int wave = threadIdx.x >> 5;
  const int tile = blockIdx.x * 8 + wave;   // 8192 tiles total
  const int b    = tile >> 10;              // batch
  const int ti   = (tile >> 5) & 31;        // row-tile
  const int tj   = tile & 31;               // col-tile

  const int r    = lane & 15;
  const int half = lane >> 4;

  // Load (x4, x5) pairs: row i = ti*16+r and col j = tj*16+r.
  // x[b,i,4] and x[b,i,5] are adjacent and 8-byte aligned.
  const float* xb = x + (size_t)b * GS * TWO_P;
  const float2 pi = *(const float2*)(xb + (ti * 16 + r) * TWO_P + 4);
  const float2 pj = *(const float2*)(xb + (tj * 16 + r) * TWO_P + 4);
  const float  ni = pi.x * pi.x + pi.y * pi.y;
  const float  nj = pj.x * pj.x + pj.y * pj.y;

  // A 16x4 f32: 2 VGPRs, lanes 0-15 -> K=0,1 ; lanes 16-31 -> K=2,3
  v2f a;
  a.x = half ? ni   : pi.x;
  a.y = half ? 1.0f : pi.y;
  // B 4x16 f32: 2 VGPRs, VGPR0 -> rows K=0 (lanes 0-15) / K=2 (lanes 16-31),
  //                      VGPR1 -> rows K=1 / K=3
  v2f bm;
  bm.x = half ? 1.0f : -2.0f * pj.x;
  bm.y = half ? nj   : -2.0f * pj.y;

  v8f c = {};
  // (neg_a, A, neg_b, B, c_mod, C, reuse_a, reuse_b)
  v8f d = __builtin_amdgcn_wmma_f32_16x16x4_f32(false, a, false, bm,
                                                (short)0, c, false, false);

  // D layout: d[v] = dsq[ti*16 + v + 8*half, tj*16 + r]
  float* orow = out + (size_t)(b * GS + ti * 16 + 8 * half) * OUTC + tj * 16 + r;
#pragma unroll
  for (int v = 0; v < 8; ++v) {
    orow[(size_t)v * OUTC] = __expf(-d[v]);   // v_exp_f32 on the trans pipe
  }
}

// ---------------------------------------------------------------------------
// Tail: out[b,i,512:522] = x[b,i,10:20]; out[b,i,522:552] = 0.
// 8*512 rows * 40 channels = 163840 threads = 640 full blocks.
// ---------------------------------------------------------------------------
__global__ __launch_bounds__(256) void tail_copy_pad(const float* __restrict__ x,
                                                     float* __restrict__ out) {
  const int t   = blockIdx.x * blockDim.x + threadIdx.x;
  const int k   = t % (P + FREE);
  const int row = t / (P + FREE);           // b*512 + i
  const float v = (k < P) ? x[(size_t)row * TWO_P + P + k] : 0.0f;
  out[(size_t)row * OUTC + GS + k] = v;
}

extern "C" void kernel_launch(void* const* d_in, const int* in_sizes, int n_in,
                              void* d_out, int out_size, void* d_ws, size_t ws_size,
                              hipStream_t stream) {
  (void)in_sizes; (void)n_in; (void)d_ws; (void)ws_size; (void)out_size;
  const float* x = (const float*)d_in[0];
  float* out = (float*)d_out;

  // 8192 tiles / 8 waves per block = 1024 blocks
  gauss_tile_wmma<<<1024, 256, 0, stream>>>(x, out);
  // 8*512*40 / 256 = 640 blocks
  tail_copy_pad<<<640, 256, 0, stream>>>(x, out);
}